// MambaRecursiveStateAdjustmentV3_41944650613242
// MI455X (gfx1250) — compile-verified
//
#include <hip/hip_runtime.h>

typedef __attribute__((ext_vector_type(2))) float v2f;
typedef __attribute__((ext_vector_type(8))) float v8f;

namespace cfg {
constexpr int C = 64, D = 128, N = 16, R = 4, K = 4, H = 32, W = 32;
constexpr int L = H * W;          // 1024
constexpr int XD = R + 2 * N;     // 36
}

__device__ __forceinline__ float silu_f(float x) { return x / (1.f + __expf(-x)); }

// ---- CDNA5 async Global->LDS helpers (ASYNCcnt path, ISA §15.18.3 / §5 flow-sync) ----
__device__ __forceinline__ void async_copy_b64(unsigned lds_off, const void* gsrc) {
  // opcode 97: GLOBAL_LOAD_ASYNC_TO_LDS_B64, GV mode (vaddr = 64-bit address)
  asm volatile("global_load_async_to_lds_b64 %0, %1, off"
               :: "v"(lds_off), "v"((unsigned long long)(uintptr_t)gsrc)
               : "memory");
}
__device__ __forceinline__ void wait_async_le2() {
  asm volatile("s_wait_asynccnt 0x2" ::: "memory");
}
__device__ __forceinline__ void wait_async_le0() {
  asm volatile("s_wait_asynccnt 0x0" ::: "memory");
}

// ---------------- LayerNorm over channel dim at each spatial position ----------------
__global__ void ln2d_kernel(const float* __restrict__ x, const float* __restrict__ g,
                            const float* __restrict__ b, float* __restrict__ out, int Cch) {
  int pos = blockIdx.x * blockDim.x + threadIdx.x;
  if (pos >= cfg::L) return;
  float mean = 0.f;
  for (int c = 0; c < Cch; ++c) mean += x[c * cfg::L + pos];
  mean /= (float)Cch;
  float var = 0.f;
  for (int c = 0; c < Cch; ++c) { float d = x[c * cfg::L + pos] - mean; var += d * d; }
  var /= (float)Cch;
  float inv = rsqrtf(var + 1e-6f);
  for (int c = 0; c < Cch; ++c)
    out[c * cfg::L + pos] = (x[c * cfg::L + pos] - mean) * inv * g[c] + b[c];
}

// ---------------- Depthwise 3x3 conv, SAME padding; act: 0=none 1=relu 2=silu --------
__global__ void dwconv3_kernel(const float* __restrict__ x, const float* __restrict__ w,
                               const float* __restrict__ b, float* __restrict__ out,
                               int Cch, int act) {
  int idx = blockIdx.x * blockDim.x + threadIdx.x;
  if (idx >= Cch * cfg::L) return;
  int c = idx / cfg::L;
  int pos = idx - c * cfg::L;
  int h = pos / cfg::W, wc = pos - h * cfg::W;
  float s = b[c];
#pragma unroll
  for (int dh = -1; dh <= 1; ++dh)
#pragma unroll
    for (int dw = -1; dw <= 1; ++dw) {
      int hh = h + dh, ww = wc + dw;
      if (hh >= 0 && hh < cfg::H && ww >= 0 && ww < cfg::W)
        s += x[c * cfg::L + hh * cfg::W + ww] * w[c * 9 + (dh + 1) * 3 + (dw + 1)];
    }
  if (act == 1) s = fmaxf(s, 0.f);
  else if (act == 2) s = silu_f(s);
  out[idx] = s;
}

// ---------------- cal_kl: exp(lp) * (lp - lq) with channel-dim log_softmax -----------
__global__ void kl_kernel(const float* __restrict__ p, const float* __restrict__ q,
                          float* __restrict__ out) {
  int pos = blockIdx.x * blockDim.x + threadIdx.x;
  if (pos >= cfg::L) return;
  float mp = -1e30f, mq = -1e30f;
  for (int c = 0; c < cfg::C; ++c) {
    mp = fmaxf(mp, p[c * cfg::L + pos]);
    mq = fmaxf(mq, q[c * cfg::L + pos]);
  }
  float sp = 0.f, sq = 0.f;
  for (int c = 0; c < cfg::C; ++c) {
    sp += __expf(p[c * cfg::L + pos] - mp);
    sq += __expf(q[c * cfg::L + pos] - mq);
  }
  float lse_p = mp + __logf(sp), lse_q = mq + __logf(sq);
  for (int c = 0; c < cfg::C; ++c) {
    float lp = p[c * cfg::L + pos] - lse_p;
    float lq = q[c * cfg::L + pos] - lse_q;
    out[c * cfg::L + pos] = __expf(lp) * (lp - lq);
  }
}

// ---------------- fp32 GEMM via V_WMMA_F32_16X16X4_F32 -------------------------------
// Y[M x Lcols] = Wm[M x Kdim] @ X[Kdim x Lcols]  (+ per-row bias if non-null)
// One wave computes a 16x64 output tile using 4 WMMA accumulators: per k-step a
// single 8-byte A load feeds 4 back-to-back v_wmma ops.
__device__ __forceinline__ void wmma_store_tile(float* __restrict__ Y, const v8f& acc,
                                                int mBase, int nB, int M, int Lcols,
                                                const float* __restrict__ bias) {
#pragma unroll
  for (int v = 0; v < 8; ++v) {
    int m = mBase + v;
    if (m < M) {
      float val = acc[v];
      if (bias) val += bias[m];
      Y[m * Lcols + nB] = val;
    }
  }
}

__global__ __launch_bounds__(32) void gemm_wmma_f32(
    const float* __restrict__ Wm, const float* __restrict__ X,
    const float* __restrict__ bias, float* __restrict__ Y,
    int M, int Kdim, int Lcols) {
  int tile = blockIdx.x;
  int tilesL = Lcols >> 6;             // 64-wide N tiles
  int m0 = (tile / tilesL) * 16;
  int n0 = (tile % tilesL) * 64;
  int lane = threadIdx.x;
  int mA = m0 + (lane & 15);
  int kHalf = (lane >> 4) * 2;         // 0 or 2
  int nB = n0 + (lane & 15);
  int kB = (lane >> 4);                // 0 or 1
  int mLoad = (mA < M) ? mA : 0;       // clamp instead of predicate
  float mMask = (mA < M) ? 1.f : 0.f;

  v8f acc0 = {}, acc1 = {}, acc2 = {}, acc3 = {};
  for (int k0 = 0; k0 < Kdim; k0 += 4) {
    float2 araw = *(const float2*)(Wm + (size_t)mLoad * Kdim + k0 + kHalf);  // b64 load
    v2f a; a.x = araw.x * mMask; a.y = araw.y * mMask;
    const float* Xk  = X + (size_t)(k0 + kB) * Lcols + nB;
    const float* Xk2 = X + (size_t)(k0 + kB + 2) * Lcols + nB;
    v2f b0; b0.x = Xk[0];  b0.y = Xk2[0];
    v2f b1; b1.x = Xk[16]; b1.y = Xk2[16];
    v2f b2; b2.x = Xk[32]; b2.y = Xk2[32];
    v2f b3; b3.x = Xk[48]; b3.y = Xk2[48];
    acc0 = __builtin_amdgcn_wmma_f32_16x16x4_f32(false, a, false, b0, (short)0, acc0, false, false);
    acc1 = __builtin_amdgcn_wmma_f32_16x16x4_f32(false, a, false, b1, (short)0, acc1, false, false);
    acc2 = __builtin_amdgcn_wmma_f32_16x16x4_f32(false, a, false, b2, (short)0, acc2, false, false);
    acc3 = __builtin_amdgcn_wmma_f32_16x16x4_f32(false, a, false, b3, (short)0, acc3, false, false);
  }
  int mBase = m0 + ((lane >> 4) * 8);
  wmma_store_tile(Y, acc0, mBase, nB,      M, Lcols, bias);
  wmma_store_tile(Y, acc1, mBase, nB + 16, M, Lcols, bias);
  wmma_store_tile(Y, acc2, mBase, nB + 32, M, Lcols, bias);
  wmma_store_tile(Y, acc3, mBase, nB + 48, M, Lcols, bias);
}

// ---------------- build the 4-direction stack xs from xin ----------------------------
__global__ void make_xs_kernel(const float* __restrict__ xin, float* __restrict__ xs) {
  int idx = blockIdx.x * blockDim.x + threadIdx.x;
  if (idx >= cfg::D * cfg::L) return;
  int d = idx / cfg::L;
  int l = idx - d * cfg::L;
  int h = l / cfg::W, w = l - h * cfg::W;
  int lt = w * cfg::H + h;
  float v = xin[idx];
  xs[(0 * cfg::D + d) * cfg::L + l] = v;
  xs[(1 * cfg::D + d) * cfg::L + lt] = v;
  xs[(2 * cfg::D + d) * cfg::L + (cfg::L - 1 - l)] = v;
  xs[(3 * cfg::D + d) * cfg::L + (cfg::L - 1 - lt)] = v;
}

__global__ void softplus_kernel(float* __restrict__ x, int n) {
  int i = blockIdx.x * blockDim.x + threadIdx.x;
  if (i >= n) return;
  float v = x[i];
  x[i] = (v > 20.f) ? v : log1pf(__expf(v));
}

// ---------------- selective scan: one (k,d) channel per 16-lane half-wave ------------
// dt/xs streams are the latency-critical sequential reads -> double-buffered async
// Global->LDS chunks (32 elements per channel) overlapped with compute (ASYNCcnt).
// Bs/Cs per-state streams stay as direct loads + global_prefetch_b8.
__global__ __launch_bounds__(32) void scan_kernel(
    const float* __restrict__ dt, const float* __restrict__ x_dbl,
    const float* __restrict__ xs, const float* __restrict__ A_log,
    const float* __restrict__ Dp, float* __restrict__ ys) {
  __shared__ float sdt[2][2][32];   // [buffer][channel][element]
  __shared__ float sxs[2][2][32];

  int lane = threadIdx.x;
  int half = lane >> 4;                      // channel within block
  int pair = blockIdx.x * 2 + half;          // global (k*D + d)
  int k = pair / cfg::D;
  int d = pair - k * cfg::D;
  int n = lane & 15;

  const float* Bs = x_dbl + (size_t)(k * cfg::XD + cfg::R + n) * cfg::L;
  const float* Cs = x_dbl + (size_t)(k * cfg::XD + cfg::R + cfg::N + n) * cfg::L;
  float* yp = ys + (size_t)pair * cfg::L;
  float A = -__expf(A_log[pair * cfg::N + n]);
  float Dv = Dp[pair];

  // per-lane async source: lane covers (channel = lane>>4, 2 consecutive elements)
  int e0 = (lane & 15) * 2;
  const float* dtSrc = dt + (size_t)(blockIdx.x * 2 + half) * cfg::L + e0;
  const float* xsSrc = xs + (size_t)(blockIdx.x * 2 + half) * cfg::L + e0;
  unsigned ldsDt0 = (unsigned)(uintptr_t)&sdt[0][half][e0];
  unsigned ldsDt1 = (unsigned)(uintptr_t)&sdt[1][half][e0];
  unsigned ldsXs0 = (unsigned)(uintptr_t)&sxs[0][half][e0];
  unsigned ldsXs1 = (unsigned)(uintptr_t)&sxs[1][half][e0];

  // prologue: stage chunk 0
  async_copy_b64(ldsDt0, dtSrc);
  async_copy_b64(ldsXs0, xsSrc);

  float h = 0.f;
  constexpr int CHUNK = 32;
  constexpr int NCHUNK = cfg::L / CHUNK;     // 32
  for (int c = 0; c < NCHUNK; ++c) {
    int buf = c & 1;
    if (c + 1 < NCHUNK) {
      // stage next chunk into the other buffer, then wait for current chunk
      async_copy_b64(buf ? ldsDt0 : ldsDt1, dtSrc + (c + 1) * CHUNK);
      async_copy_b64(buf ? ldsXs0 : ldsXs1, xsSrc + (c + 1) * CHUNK);
      wait_async_le2();                      // current chunk's 2 transfers done
    } else {
      wait_async_le0();
    }
    int lBase = c * CHUNK;
    if (lBase + CHUNK < cfg::L) {
      __builtin_prefetch(Bs + lBase + CHUNK, 0, 1);   // global_prefetch_b8
      __builtin_prefetch(Cs + lBase + CHUNK, 0, 1);
    }
    for (int j = 0; j < CHUNK; ++j) {
      int l = lBase + j;
      float dtv = sdt[buf][half][j];
      float xv = sxs[buf][half][j];
      float a = __expf(dtv * A);
      float bu = dtv * Bs[l] * xv;
      h = a * h + bu;
      float contrib = h * Cs[l];
      // reduce over the 16 state lanes (stays within each half-wave)
      contrib += __shfl_xor(contrib, 1, 32);
      contrib += __shfl_xor(contrib, 2, 32);
      contrib += __shfl_xor(contrib, 4, 32);
      contrib += __shfl_xor(contrib, 8, 32);
      if (n == 0) yp[l] = contrib + Dv * xv;
    }
  }
}

// ---------------- merge 4 scan directions back to (D, H, W) --------------------------
__global__ void combine_ys_kernel(const float* __restrict__ ys, float* __restrict__ y) {
  int idx = blockIdx.x * blockDim.x + threadIdx.x;
  if (idx >= cfg::D * cfg::L) return;
  int d = idx / cfg::L;
  int l = idx - d * cfg::L;
  int h = l / cfg::W, w = l - h * cfg::W;
  int lt = w * cfg::H + h;
  float v = ys[(0 * cfg::D + d) * cfg::L + l]
          + ys[(1 * cfg::D + d) * cfg::L + lt]
          + ys[(2 * cfg::D + d) * cfg::L + (cfg::L - 1 - l)]
          + ys[(3 * cfg::D + d) * cfg::L + (cfg::L - 1 - lt)];
  y[idx] = v;
}

__global__ void gate_kernel(float* __restrict__ y, const float* __restrict__ z, int n) {
  int i = blockIdx.x * blockDim.x + threadIdx.x;
  if (i < n) y[i] *= silu_f(z[i]);
}

__global__ void cur_init_kernel(const float* __restrict__ prev, const float* __restrict__ s0,
                                const float* __restrict__ bias, float* __restrict__ cur, int n) {
  int i = blockIdx.x * blockDim.x + threadIdx.x;
  if (i < n) cur[i] = prev[i] * __expf(-s0[i]) + bias[i];
}

__global__ void cur_update_kernel(const float* __restrict__ cur, const float* __restrict__ ss,
                                  const float* __restrict__ si, const float* __restrict__ bstep,
                                  float* __restrict__ curN, int n) {
  int i = blockIdx.x * blockDim.x + threadIdx.x;
  // cur / (exp(-ss)*exp(-si)) + bstep == cur * exp(ss+si) + bstep
  if (i < n) curN[i] = cur[i] * __expf(ss[i] + si[i]) + bstep[i];
}

// =====================================================================================
// Host side
// =====================================================================================
struct MambaParams {
  const float *in_proj_w, *conv_w, *conv_b, *x_proj_w, *dt_proj_w, *dt_proj_b,
              *A_log, *Dp, *out_norm_g, *out_norm_b, *out_proj_w;
};

static MambaParams load_mamba(void* const* d_in, int base) {
  MambaParams m;
  m.in_proj_w  = (const float*)d_in[base + 0];
  m.conv_w     = (const float*)d_in[base + 1];
  m.conv_b     = (const float*)d_in[base + 2];
  m.x_proj_w   = (const float*)d_in[base + 3];
  m.dt_proj_w  = (const float*)d_in[base + 4];
  m.dt_proj_b  = (const float*)d_in[base + 5];
  m.A_log      = (const float*)d_in[base + 6];
  m.Dp         = (const float*)d_in[base + 7];
  m.out_norm_g = (const float*)d_in[base + 8];
  m.out_norm_b = (const float*)d_in[base + 9];
  m.out_proj_w = (const float*)d_in[base + 10];
  return m;
}

struct WS {
  float *xz, *xin, *xs, *x_dbl, *dt, *ys, *y, *y2;
};

static void run_ss2d(const float* x, const MambaParams& m, float* out, const WS& ws,
                     hipStream_t s) {
  using namespace cfg;
  const int TPB = 256;
  const int tilesN = L / 64;   // 16
  // in_proj: (2D x C) @ (C x L) -> xz
  gemm_wmma_f32<<<(2 * D / 16) * tilesN, 32, 0, s>>>(m.in_proj_w, x, nullptr, ws.xz, 2 * D, C, L);
  // depthwise conv + silu on xin slice
  dwconv3_kernel<<<(D * L + TPB - 1) / TPB, TPB, 0, s>>>(ws.xz, m.conv_w, m.conv_b, ws.xin, D, 2);
  // 4-direction stack
  make_xs_kernel<<<(D * L + TPB - 1) / TPB, TPB, 0, s>>>(ws.xin, ws.xs);
  // x_proj per direction: (XD x D) @ (D x L)
  for (int k = 0; k < K; ++k) {
    int tilesM = (XD + 15) / 16;   // 3
    gemm_wmma_f32<<<tilesM * tilesN, 32, 0, s>>>(
        m.x_proj_w + (size_t)k * XD * D, ws.xs + (size_t)k * D * L, nullptr,
        ws.x_dbl + (size_t)k * XD * L, XD, D, L);
  }
  // dt = dt_proj_w @ dts + dt_proj_b : (D x R) @ (R x L)
  for (int k = 0; k < K; ++k) {
    gemm_wmma_f32<<<(D / 16) * tilesN, 32, 0, s>>>(
        m.dt_proj_w + (size_t)k * D * R, ws.x_dbl + (size_t)k * XD * L,
        m.dt_proj_b + (size_t)k * D, ws.dt + (size_t)k * D * L, D, R, L);
  }
  softplus_kernel<<<(K * D * L + TPB - 1) / TPB, TPB, 0, s>>>(ws.dt, K * D * L);
  // selective scan
  scan_kernel<<<K * D / 2, 32, 0, s>>>(ws.dt, ws.x_dbl, ws.xs, m.A_log, m.Dp, ws.ys);
  // combine directions, out_norm, gate, out_proj
  combine_ys_kernel<<<(D * L + TPB - 1) / TPB, TPB, 0, s>>>(ws.ys, ws.y);
  ln2d_kernel<<<(L + TPB - 1) / TPB, TPB, 0, s>>>(ws.y, m.out_norm_g, m.out_norm_b, ws.y2, D);
  gate_kernel<<<(D * L + TPB - 1) / TPB, TPB, 0, s>>>(ws.y2, ws.xz + (size_t)D * L, D * L);
  gemm_wmma_f32<<<(C / 16) * tilesN, 32, 0, s>>>(m.out_proj_w, ws.y2, nullptr, out, C, D, L);
}

extern "C" void kernel_launch(void* const* d_in, const int* in_sizes, int n_in,
                              void* d_out, int out_size, void* d_ws, size_t ws_size,
                              hipStream_t stream) {
  (void)in_sizes; (void)n_in; (void)out_size; (void)ws_size;
  using namespace cfg;
  const float* image_sequence = (const float*)d_in[0];
  const float* difficult_zone = (const float*)d_in[1];
  // d_in[2] = sigma (unused by the reference forward)
  const float* norm_dz_g = (const float*)d_in[3];
  const float* norm_dz_b = (const float*)d_in[4];
  const float* norm_img_g = (const float*)d_in[5];
  const float* norm_img_b = (const float*)d_in[6];
  const float* conv1_w = (const float*)d_in[7];
  const float* conv1_b = (const float*)d_in[8];
  const float* conv2_w = (const float*)d_in[9];
  const float* conv2_b = (const float*)d_in[10];
  MambaParams m_state = load_mamba(d_in, 11);
  MambaParams m_image = load_mamba(d_in, 22);
  MambaParams m_bias  = load_mamba(d_in, 33);

  float* w = (float*)d_ws;
  size_t off = 0;
  auto alloc = [&](size_t n) { float* p = w + off; off += n; return p; };
  float* bufA   = alloc((size_t)C * L);   // prev (initial)
  float* bufB   = alloc((size_t)C * L);   // cur (initial)
  float* bufCx  = alloc((size_t)C * L);   // next
  float* s0     = alloc((size_t)C * L);
  float* tin    = alloc((size_t)C * L);
  float* biasO  = alloc((size_t)C * L);
  float* sigImg = alloc((size_t)C * L);
  float* sigSt  = alloc((size_t)C * L);
  float* bStep  = alloc((size_t)C * L);
  float* lnimg  = alloc((size_t)C * L);
  WS ws;
  ws.xz    = alloc((size_t)2 * D * L);
  ws.xin   = alloc((size_t)D * L);
  ws.xs    = alloc((size_t)K * D * L);
  ws.x_dbl = alloc((size_t)K * XD * L);
  ws.dt    = alloc((size_t)K * D * L);
  ws.ys    = alloc((size_t)K * D * L);
  ws.y     = alloc((size_t)D * L);
  ws.y2    = alloc((size_t)D * L);

  float* pPrev = bufA;
  float* pCur  = bufB;
  float* pNext = bufCx;

  const int TPB = 256;
  const int nCL = C * L;

  // prev = ln2d(difficult_zone)
  ln2d_kernel<<<(L + TPB - 1) / TPB, TPB, 0, stream>>>(difficult_zone, norm_dz_g, norm_dz_b, pPrev, C);
  // s0 = relu(dwconv2(relu(dwconv1(prev))))
  dwconv3_kernel<<<(nCL + TPB - 1) / TPB, TPB, 0, stream>>>(pPrev, conv1_w, conv1_b, tin, C, 1);
  dwconv3_kernel<<<(nCL + TPB - 1) / TPB, TPB, 0, stream>>>(tin, conv2_w, conv2_b, s0, C, 1);
  // bias = ss2d(prev, mamba_bias); cur = prev/exp(s0) + bias
  run_ss2d(pPrev, m_bias, biasO, ws, stream);
  cur_init_kernel<<<(nCL + TPB - 1) / TPB, TPB, 0, stream>>>(pPrev, s0, biasO, pCur, nCL);

  for (int i = 0; i < 2; ++i) {  // LENGTH = 2
    const float* img = image_sequence + (size_t)i * C * L;
    ln2d_kernel<<<(L + TPB - 1) / TPB, TPB, 0, stream>>>(img, norm_img_g, norm_img_b, lnimg, C);
    run_ss2d(lnimg, m_image, sigImg, ws, stream);
    kl_kernel<<<(L + TPB - 1) / TPB, TPB, 0, stream>>>(pCur, pPrev, tin);
    run_ss2d(tin, m_state, sigSt, ws, stream);
    run_ss2d(pCur, m_bias, bStep, ws, stream);
    cur_update_kernel<<<(nCL + TPB - 1) / TPB, TPB, 0, stream>>>(pCur, sigSt, sigImg, bStep, pNext, nCL);
    float* t = pPrev; pPrev = pCur; pCur = pNext; pNext = t;
  }

  hipMemcpyAsync(d_out, pCur, (size_t)nCL * sizeof(float), hipMemcpyDeviceToDevice, stream);
}